// GCNEncoder_42640435315022
// MI455X (gfx1250) — compile-verified
//
#include <hip/hip_runtime.h>
#include <hip/hip_bf16.h>
#include <stdint.h>

// ---------------------------------------------------------------------------
// 2-layer GCN encoder for MI455X (gfx1250, wave32).
//  Pipeline per call:
//   1) deg[i] = 1 + in-degree(i)  (self loops), dinv = rsqrt(deg)   [once]
//   2) layer: xw = A @ W  (WMMA f32 16x16x4, fp32 accumulate)
//             agg[i,:] = xw[i,:] * dinv[i]^2              (self-loop msg)
//             agg[d,:] += xw[s,:] * dinv[s]*dinv[d]       (edge scatter,
//                                                          global f32 atomics,
//                                                          one wave per edge,
//                                                          float4 per lane)
//             out[:,colOff:colOff+128] = relu(agg + b)
//  Layer 2 reads h1 back from d_out (row stride 256), writes cols 128..255.
// ---------------------------------------------------------------------------

#define IN_CH 128
#define HID   128

typedef __attribute__((ext_vector_type(2))) float v2f;
typedef __attribute__((ext_vector_type(8))) float v8f;

// ---------------- degree / norm kernels ----------------

__global__ void k_deg_init(float* __restrict__ deg, int n) {
    int i = blockIdx.x * blockDim.x + threadIdx.x;
    if (i < n) deg[i] = 1.0f;            // self loop contributes 1
}

__global__ void k_deg_acc(const int* __restrict__ dst, float* __restrict__ deg, int e) {
    int i = blockIdx.x * blockDim.x + threadIdx.x;
    if (i < e) atomicAdd(&deg[dst[i]], 1.0f);
}

__global__ void k_deg_rsqrt(float* __restrict__ deg, int n) {
    int i = blockIdx.x * blockDim.x + threadIdx.x;
    if (i < n) deg[i] = rsqrtf(deg[i]);  // deg >= 1 always (self loop)
}

// ---------------- WMMA GEMM:  out[M,128] = A[M,K=128] (row stride lda) @ W[128,128]
// One wave computes one 16x16 output tile. fp32 WMMA, K stepped by 4.

__global__ void k_gemm_wmma(const float* __restrict__ A, int lda,
                            const float* __restrict__ W,
                            float* __restrict__ out, int M) {
    const int wave = threadIdx.x >> 5;          // 4 waves / block
    const int lane = threadIdx.x & 31;
    const int mt   = blockIdx.x;                // row tile
    const int nt   = blockIdx.y * 4 + wave;     // col tile 0..7
    const int row0 = mt * 16;
    const int col0 = nt * 16;

    const int ln = lane & 15;                   // 0..15
    const int kb = (lane >> 4) * 2;             // 0 or 2 (K sub-offset)

    // clamp row for loads so EXEC stays all-ones around WMMA
    const int arow = (row0 + ln < M) ? (row0 + ln) : (M - 1);

    v8f c = {};
    #pragma unroll 4
    for (int k = 0; k < IN_CH; k += 4) {
        // A fragment: lane holds row M=ln, K = k+kb (vgpr0), k+kb+1 (vgpr1)
        const float* ap = A + (size_t)arow * lda + (k + kb);
        v2f a; a.x = ap[0]; a.y = ap[1];
        // B fragment: lane holds col N=ln, K = k+kb (vgpr0), k+kb+1 (vgpr1)
        v2f b;
        b.x = W[(size_t)(k + kb + 0) * HID + col0 + ln];
        b.y = W[(size_t)(k + kb + 1) * HID + col0 + ln];
        c = __builtin_amdgcn_wmma_f32_16x16x4_f32(
                /*neg_a=*/false, a, /*neg_b=*/false, b,
                /*c_mod=*/(short)0, c, /*reuse_a=*/false, /*reuse_b=*/false);
    }

    // D layout: vgpr i -> rows (lane>>4)*8 + i, col = lane&15
    const int mb = (lane >> 4) * 8;
    float* op = out + (size_t)(row0 + mb) * HID + col0 + ln;
    if (row0 + 16 <= M) {
        // fast path: whole tile in range (always taken when M % 16 == 0)
        #pragma unroll
        for (int i = 0; i < 8; ++i) op[(size_t)i * HID] = c[i];
    } else {
        #pragma unroll
        for (int i = 0; i < 8; ++i) {
            if (row0 + mb + i < M) op[(size_t)i * HID] = c[i];
        }
    }
}

// ---------------- self-loop init: agg = xw * dinv^2 ----------------

__global__ void k_selfloop_init(const float* __restrict__ xw,
                                const float* __restrict__ dinv,
                                float* __restrict__ agg, int n) {
    long long i = (long long)blockIdx.x * blockDim.x + threadIdx.x;  // over n*128
    if (i >= (long long)n * HID) return;
    int node = (int)(i >> 7);
    float dv = dinv[node];
    agg[i] = xw[i] * dv * dv;
}

// ---------------- edge scatter: one wave per edge ----------------
// Lane l handles 4 consecutive floats -> whole wave moves one 512B row.

__global__ void k_edge_scatter(const int* __restrict__ src,
                               const int* __restrict__ dst,
                               const float* __restrict__ dinv,
                               const float* __restrict__ xw,
                               float* __restrict__ agg, int e) {
    long long t   = (long long)blockIdx.x * blockDim.x + threadIdx.x;
    int       eid = (int)(t >> 5);
    int       lane = threadIdx.x & 31;
    if (eid >= e) return;
    int s = src[eid];
    int d = dst[eid];
    float norm = dinv[s] * dinv[d];
    const float4* xp = (const float4*)(xw + (size_t)s * HID);
    float4 v = xp[lane];                       // 16B per lane, 512B per wave
    float* ap = agg + (size_t)d * HID + lane * 4;
    atomicAdd(ap + 0, v.x * norm);
    atomicAdd(ap + 1, v.y * norm);
    atomicAdd(ap + 2, v.z * norm);
    atomicAdd(ap + 3, v.w * norm);
}

// ---------------- bias + relu, write into d_out column block ----------------

__global__ void k_bias_relu(const float* __restrict__ agg,
                            const float* __restrict__ bias,
                            float* __restrict__ out, int colOff, int ldo, int n) {
    long long i = (long long)blockIdx.x * blockDim.x + threadIdx.x;  // over n*128
    if (i >= (long long)n * HID) return;
    int node = (int)(i >> 7);
    int c    = (int)(i & (HID - 1));
    float v = agg[i] + bias[c];
    out[(size_t)node * ldo + colOff + c] = fmaxf(v, 0.0f);
}

// ---------------------------------------------------------------------------

extern "C" void kernel_launch(void* const* d_in, const int* in_sizes, int n_in,
                              void* d_out, int out_size, void* d_ws, size_t ws_size,
                              hipStream_t stream) {
    const float* x  = (const float*)d_in[0];
    const int*   ei = (const int*)  d_in[1];
    const float* W1 = (const float*)d_in[2];
    const float* b1 = (const float*)d_in[3];
    const float* W2 = (const float*)d_in[4];
    const float* b2 = (const float*)d_in[5];
    float* out = (float*)d_out;

    const int N = in_sizes[0] / IN_CH;      // 100000
    const int E = in_sizes[1] / 2;          // 3200000
    const int* src = ei;
    const int* dst = ei + E;

    // workspace layout: dinv | xw | agg
    char*  ws   = (char*)d_ws;
    float* dinv = (float*)ws;
    size_t off  = ((size_t)N * sizeof(float) + 255) & ~(size_t)255;
    float* xw   = (float*)(ws + off);
    size_t xwB  = (size_t)N * HID * sizeof(float);
    float* agg  = (float*)(ws + off + xwB);

    const int TB = 256;
    const long long NH = (long long)N * HID;
    const int gridNH   = (int)((NH + TB - 1) / TB);
    const int gridN    = (N + TB - 1) / TB;
    const int gridE    = (E + TB - 1) / TB;
    const int gridEw   = (int)(((long long)E * 32 + TB - 1) / TB);  // wave per edge
    const int mTiles   = (N + 15) / 16;
    dim3 gGemm(mTiles, 2);   // 2 * 4 waves = 8 column tiles of 16 (HID=128)
    dim3 bGemm(128);

    // ---- degrees (shared by both layers) ----
    k_deg_init <<<gridN, TB, 0, stream>>>(dinv, N);
    k_deg_acc  <<<gridE, TB, 0, stream>>>(dst, dinv, E);
    k_deg_rsqrt<<<gridN, TB, 0, stream>>>(dinv, N);

    // ---- layer 1 ----
    k_gemm_wmma    <<<gGemm, bGemm, 0, stream>>>(x, IN_CH, W1, xw, N);
    k_selfloop_init<<<gridNH, TB, 0, stream>>>(xw, dinv, agg, N);
    k_edge_scatter <<<gridEw, TB, 0, stream>>>(src, dst, dinv, xw, agg, E);
    k_bias_relu    <<<gridNH, TB, 0, stream>>>(agg, b1, out, 0, 2 * HID, N);

    // ---- layer 2 (input h1 read from d_out, row stride 256) ----
    k_gemm_wmma    <<<gGemm, bGemm, 0, stream>>>(out, 2 * HID, W2, xw, N);
    k_selfloop_init<<<gridNH, TB, 0, stream>>>(xw, dinv, agg, N);
    k_edge_scatter <<<gridEw, TB, 0, stream>>>(src, dst, dinv, xw, agg, E);
    k_bias_relu    <<<gridNH, TB, 0, stream>>>(agg, b2, out, HID, 2 * HID, N);
}